// STR_Loss_DTW_40845138985587
// MI455X (gfx1250) — compile-verified
//
#include <hip/hip_runtime.h>
#include <hip/hip_bf16.h>
#include <math.h>

// ---------------------------------------------------------------------------
// STR DTW loss for MI455X (gfx1250, wave32).
//  K1: distance matrix via V_WMMA_F32_16X16X4_F32 (cross term), f32 end to end
//  K2: DTW DP, one workgroup per batch; row recurrence converted to
//      sum-scan + min-scan (log-depth) using the identity
//      a_j = S_j + min_{k<=j} (m_k - S_{k-1})
//  K3: per-batch sequential backtrack (one lane each), loss accumulation
//  K4: finalize 4 scalar outputs
// Workspace: acc matrix B*256*256 f32 (64 MB, L2-resident) + 3 f32 partials.
// ---------------------------------------------------------------------------

typedef __attribute__((ext_vector_type(2))) float v2f;
typedef __attribute__((ext_vector_type(8))) float v8f;

#define S_LEN 256
#define PATH_L (2 * S_LEN - 1)
#define SOS_W 5.0f

// ---------------- K1: pairwise distances with WMMA -------------------------
__global__ __launch_bounds__(256) void dist_wmma_kernel(
    const float* __restrict__ preds, const float* __restrict__ targets,
    float* __restrict__ acc, float* __restrict__ part, int B)
{
  // zero the loss accumulators once (K3 runs strictly after this kernel)
  if (blockIdx.x == 0 && threadIdx.x < 3) part[threadIdx.x] = 0.0f;

  const int S = S_LEN;
  int wave = (int)((blockIdx.x * blockDim.x + threadIdx.x) >> 5);
  int lane = (int)(threadIdx.x & 31);
  int b    = wave >> 8;          // 256 tiles (16x16 grid of 16x16) per batch
  int tile = wave & 255;
  if (b >= B) return;            // wave-uniform; EXEC stays all-ones
  int n0 = (tile >> 4) << 4;     // pred-row tile base
  int m0 = (tile & 15) << 4;     // target-col tile base

  bool lo  = lane < 16;
  int  col = lane & 15;

  // A operand: 16x4 f32, K=0,1 = (px,py) in lanes 0-15; K=2,3 zero-padded.
  int nA = n0 + (lo ? lane : 0);
  const float* pA = preds + ((size_t)b * S + nA) * 4;
  v2f av;
  av.x = lo ? pA[0] : 0.0f;
  av.y = lo ? pA[1] : 0.0f;

  // B operand: 4x16 f32, rows K=0,1 = (tx,ty) in lanes 0-15; K=2,3 zero.
  const float* pT = targets + ((size_t)b * S + (m0 + col)) * 4;
  float tx = pT[0], ty = pT[1];
  v2f bv;
  bv.x = lo ? tx : 0.0f;
  bv.y = lo ? ty : 0.0f;

  v8f c = {0.f, 0.f, 0.f, 0.f, 0.f, 0.f, 0.f, 0.f};
  // D = A x B (+0): c[r] = dot(p[n0+r+8*hi], t[m0+col])
  c = __builtin_amdgcn_wmma_f32_16x16x4_f32(
      /*neg_a=*/false, av, /*neg_b=*/false, bv,
      /*c_mod=*/(short)0, c, /*reuse_a=*/false, /*reuse_b=*/false);

  float tm2 = tx * tx + ty * ty;
  int hi8 = lo ? 0 : 8;
#pragma unroll
  for (int r = 0; r < 8; ++r) {
    int n = n0 + r + hi8;
    const float* pr = preds + ((size_t)b * S + n) * 4;
    float px = pr[0], py = pr[1];
    float d2 = px * px + py * py + tm2 - 2.0f * c[r];
    d2 = fmaxf(d2, 0.0f) + 1e-12f;
    acc[((size_t)b * S + n) * S + (m0 + col)] = sqrtf(d2);
  }
}

// ---------------- K2: DTW accumulation (scan formulation) ------------------
__device__ inline float lds_scan_add(float* buf, int j, float v) {
  buf[j] = v;
  __syncthreads();
#pragma unroll
  for (int off = 1; off < S_LEN; off <<= 1) {
    float t = buf[j];
    float u = (j >= off) ? buf[j - off] : 0.0f;
    __syncthreads();
    buf[j] = t + u;
    __syncthreads();
  }
  return buf[j];
}

__device__ inline float lds_scan_min(float* buf, int j, float v) {
  buf[j] = v;
  __syncthreads();
#pragma unroll
  for (int off = 1; off < S_LEN; off <<= 1) {
    float t = buf[j];
    float u = (j >= off) ? buf[j - off] : INFINITY;
    __syncthreads();
    buf[j] = fminf(t, u);
    __syncthreads();
  }
  return buf[j];
}

__global__ __launch_bounds__(S_LEN) void dtw_dp_kernel(float* __restrict__ acc)
{
  const int S = S_LEN;
  int b = blockIdx.x;
  int j = threadIdx.x;
  float* A = acc + (size_t)b * S * S;

  __shared__ float prev[S_LEN];
  __shared__ float buf[S_LEN];

  // row 0: acc[0][j] = cumsum(D[0][:j])
  float a = lds_scan_add(buf, j, A[j]);
  A[j] = a;
  prev[j] = a;
  __syncthreads();

  for (int i = 1; i < S; ++i) {
    float m = fminf(prev[j], (j > 0) ? prev[j - 1] : INFINITY);
    float d = A[(size_t)i * S + j];
    __syncthreads();                       // prev reads done before buf reuse

    float Sj = lds_scan_add(buf, j, d);    // inclusive prefix sum of d
    float Sjm1 = (j > 0) ? buf[j - 1] : 0.0f;
    __syncthreads();

    float P = lds_scan_min(buf, j, m - Sjm1);  // prefix min of (m_k - S_{k-1})
    a = Sj + P;                            // a_j = d_j + min(m_j, a_{j-1})
    A[(size_t)i * S + j] = a;
    prev[j] = a;
    __syncthreads();
  }
}

// ---------------- K3: backtrack + losses -----------------------------------
__device__ inline float softplusf(float x) {
  return fmaxf(x, 0.0f) + log1pf(expf(-fabsf(x)));
}
__device__ inline float bce_logits(float x, float t, float w) {
  return w * t * softplusf(-x) + (1.0f - t) * softplusf(x);
}

__global__ __launch_bounds__(32) void backtrack_kernel(
    const float* __restrict__ preds, const float* __restrict__ targets,
    const float* __restrict__ acc, float* __restrict__ part, int B)
{
  const int S = S_LEN;
  int b = (int)(blockIdx.x * 32 + threadIdx.x);
  if (b >= B) return;

  const float* A = acc + (size_t)b * S * S;
  const float* P = preds + (size_t)b * S * 4;
  const float* T = targets + (size_t)b * S * 4;

  int i = S - 1, j = S - 1;
  bool done = false;
  float sd = 0.0f, ss = 0.0f, se = 0.0f, cnt = 0.0f;

  for (int step = 0; step < PATH_L; ++step) {
    if (!done) {
      const float* pp = P + (size_t)i * 4;
      const float* tt = T + (size_t)j * 4;
      float dx = pp[0] - tt[0], dy = pp[1] - tt[1];
      sd += sqrtf(dx * dx + dy * dy + 1e-12f);
      ss += bce_logits(pp[2], tt[2], SOS_W);
      se += bce_logits(pp[3], tt[3], 1.0f);
      cnt += 1.0f;

      if (i == 0 && j == 0) {
        done = true;
      } else {
        float diag = (i > 0 && j > 0) ? A[(size_t)(i - 1) * S + (j - 1)] : INFINITY;
        float up   = (i > 0)          ? A[(size_t)(i - 1) * S + j]       : INFINITY;
        float left = (j > 0)          ? A[(size_t)i * S + (j - 1)]       : INFINITY;
        // argmin over [diag, up, left]; first-min wins (diag preferred)
        int k = 0;
        float best = diag;
        if (up < best)   { best = up;   k = 1; }
        if (left < best) { best = left; k = 2; }
        i += (k == 2) ? 0 : -1;
        j += (k == 1) ? 0 : -1;
      }
    }
  }
  float invB = 1.0f / (float)B;
  atomicAdd(&part[0], (sd / cnt) * invB);
  atomicAdd(&part[1], (ss / cnt) * invB);
  atomicAdd(&part[2], (se / cnt) * invB);
}

// ---------------- K4: finalize ---------------------------------------------
__global__ void finalize_kernel(const float* __restrict__ part, float* __restrict__ out)
{
  if (threadIdx.x == 0 && blockIdx.x == 0) {
    float d = part[0], s = part[1], e = part[2];
    out[0] = d + s + e;
    out[1] = d;
    out[2] = s;
    out[3] = e;
  }
}

// ---------------------------------------------------------------------------
extern "C" void kernel_launch(void* const* d_in, const int* in_sizes, int n_in,
                              void* d_out, int out_size, void* d_ws, size_t ws_size,
                              hipStream_t stream)
{
  (void)n_in; (void)out_size;
  const float* preds   = (const float*)d_in[0];
  const float* targets = (const float*)d_in[1];
  int B = in_sizes[0] / (S_LEN * 4);        // (B, 256, 4) f32

  size_t acc_elems = (size_t)B * S_LEN * S_LEN;
  if (ws_size < (acc_elems + 3) * sizeof(float)) return;  // workspace too small

  float* acc  = (float*)d_ws;
  float* part = acc + acc_elems;
  float* out  = (float*)d_out;

  // K1: one wave per 16x16 tile; 256 tiles/batch; 8 waves per 256-thread block
  int waves   = B * 256;
  int blocks1 = waves / 8;
  dist_wmma_kernel<<<blocks1, 256, 0, stream>>>(preds, targets, acc, part, B);

  // K2: one workgroup per batch
  dtw_dp_kernel<<<B, S_LEN, 0, stream>>>(acc);

  // K3: one lane per batch
  backtrack_kernel<<<(B + 31) / 32, 32, 0, stream>>>(preds, targets, acc, part, B);

  // K4
  finalize_kernel<<<1, 1, 0, stream>>>(part, out);
}